// CausalGrassmannMixer_60438779789931
// MI455X (gfx1250) — compile-verified
//
#include <hip/hip_runtime.h>
#include <math.h>

typedef float v2f __attribute__((ext_vector_type(2)));
typedef float v8f __attribute__((ext_vector_type(8)));
typedef unsigned int u32x4 __attribute__((ext_vector_type(4)));
typedef int i32x4 __attribute__((ext_vector_type(4)));
typedef int i32x8 __attribute__((ext_vector_type(8)));

// D = A(16x4 f32) * B(4x16 f32) + C(16x16 f32)
#define WMMA_F32(a, b, c) \
  __builtin_amdgcn_wmma_f32_16x16x4_f32(false, (a), false, (b), (short)0, (c), false, false)

__device__ __forceinline__ float gelu_exact(float x) {
  return 0.5f * x * (1.0f + erff(x * 0.70710678118654752f));
}

// ---------------------------------------------------------------------------
// TDM: load a 64x64 f32 tile (global row stride 1024 floats) into LDS with
// 4-DWORD padding after every 64 DWORDs -> LDS row stride 68 floats.
// D# per CDNA5 ISA ch.8: group0 {count, lds_addr, global_addr, type=2},
// group1 {data_size=4B, pad_enable, pad_interval=64DW, pad_amount=4DW,
//         tensor_dim0=1024, tensor_dim1=16384, tile 64x64, dim0_stride=1024}.
// Toolchain exposes the 6-arg builtin: (u32x4, i32x8, i32x4, i32x4, i32x8, i32).
// Groups 2/3 (3D+ dims / gather indices) unused for a 2D tile -> zeros.
// ---------------------------------------------------------------------------
__device__ __forceinline__ void tdm_load_64x64(const float* gsrc, unsigned lds_off) {
  unsigned long long ga = (unsigned long long)(uintptr_t)gsrc;
  u32x4 g0;
  g0[0] = 1u;                                   // count=1 (valid), user mode
  g0[1] = lds_off;                              // lds_addr (bytes, group-rel)
  g0[2] = (unsigned)ga;                         // global_addr[31:0]
  g0[3] = ((unsigned)(ga >> 32) & 0x01FFFFFFu)  // global_addr[56:32]
          | (2u << 30);                         // type=2 ("image")
  i32x8 g1;
  g1[0] = (2 << 16)      // data_size = 4 bytes
        | (1 << 20)      // pad_enable
        | (5 << 22)      // pad_interval: 2^(5+1) = 64 DWORDs
        | (3 << 25);     // pad_amount: 3+1 = 4 DWORDs
  g1[1] = (int)(1024u << 16);   // tensor_dim0 = 1024 (bits 79:48, low half)
  g1[2] = (int)(16384u << 16);  // tensor_dim1 = 16384 (bits 111:80, low half)
  g1[3] = (int)(64u << 16);     // tile_dim0 = 64 (bits 127:112)
  g1[4] = 64;                   // tile_dim1 = 64 (bits 143:128); tile_dim2=0
  g1[5] = 1024;                 // tensor_dim0_stride = 1024 (bits 207:160)
  g1[6] = 0;
  g1[7] = 0;
  i32x4 zz4 = {0, 0, 0, 0};
  i32x8 zz8 = {0, 0, 0, 0, 0, 0, 0, 0};
  __builtin_amdgcn_tensor_load_to_lds(g0, g1, zz4, zz4, zz8, 0);
}

// ---------------------------------------------------------------------------
// Kernel 1: z = h @ red_w + red_b   (16384 x 1024) @ (1024 x 16)
// ---------------------------------------------------------------------------
__global__ __launch_bounds__(256) void zk(const float* __restrict__ h,
                                          const float* __restrict__ rw,
                                          const float* __restrict__ rb,
                                          float* __restrict__ z, int ntok) {
  int wid  = (blockIdx.x * blockDim.x + threadIdx.x) >> 5;
  int lane = threadIdx.x & 31;
  if (wid >= ntok) return;
  const float* hr = h + (size_t)wid * 1024;
  float acc[16];
#pragma unroll
  for (int r = 0; r < 16; r++) acc[r] = 0.f;
  for (int k = lane; k < 1024; k += 32) {
    float hv = hr[k];
    const float4* w4 = (const float4*)(rw + k * 16);
#pragma unroll
    for (int q = 0; q < 4; q++) {
      float4 w = w4[q];
      acc[q * 4 + 0] += hv * w.x;
      acc[q * 4 + 1] += hv * w.y;
      acc[q * 4 + 2] += hv * w.z;
      acc[q * 4 + 3] += hv * w.w;
    }
  }
#pragma unroll
  for (int r = 0; r < 16; r++) {
    float v = acc[r];
#pragma unroll
    for (int off = 16; off >= 1; off >>= 1) v += __shfl_xor(v, off, 32);
    acc[r] = v;
  }
  if (lane == 0) {
    float* zr = z + (size_t)wid * 16;
#pragma unroll
    for (int r = 0; r < 16; r++) zr[r] = acc[r] + rb[r];
  }
}

// ---------------------------------------------------------------------------
// Kernel 2: geometry path. Block = 16 tokens, 128 threads (4 waves).
// ---------------------------------------------------------------------------
#define PSTR 120
#define QSTR 258

__global__ __launch_bounds__(128) void geomk(const float* __restrict__ z,
                                             const float* __restrict__ g1w,
                                             const float* __restrict__ g1b,
                                             const float* __restrict__ g2w,
                                             const float* __restrict__ g2b,
                                             float* __restrict__ g) {
  __shared__ float P[96 * PSTR];
  __shared__ float ZQ[16 * QSTR];  // zbuf (phases 1-2), then qbuf (3-4)
  const int tid  = threadIdx.x;
  const int lane = tid & 31;
  const int wid  = tid >> 5;
  const int ln   = lane & 15;
  const int hi   = lane >> 4;
  const int koff = hi * 2;
  const int t0 = blockIdx.x * 16;
  const int l0 = t0 & 4095;

  // phase 1: stage z rows [l0-32 .. l0+15]
  float* zb = ZQ;
  if (tid < 48) {
    int li = l0 + tid - 32;
    float* dst = zb + tid * 16;
    if (li >= 0) {
      const float* src = z + (size_t)(t0 + tid - 32) * 16;
#pragma unroll
      for (int r = 0; r < 16; r++) dst[r] = src[r];
    } else {
#pragma unroll
      for (int r = 0; r < 16; r++) dst[r] = 0.f;
    }
  }
  __syncthreads();

  // phase 2: Plucker rows (row = d*16 + m, delta = 1<<d), two-pass norm
  if (tid < 96) {
    int d = tid >> 4, m = tid & 15;
    int delta = 1 << d;
    float u[16], v[16];
    const float* ur = zb + (32 + m - delta) * 16;
    const float* vr = zb + (32 + m) * 16;
#pragma unroll
    for (int r = 0; r < 16; r++) { u[r] = ur[r]; v[r] = vr[r]; }
    float ss = 0.f;
#pragma unroll
    for (int i = 0; i < 16; i++)
#pragma unroll
      for (int j = i + 1; j < 16; j++) {
        float p = u[i] * v[j] - u[j] * v[i];
        ss += p * p;
      }
    float inv = 1.0f / fmaxf(sqrtf(ss), 1e-6f);
    float* pr = P + tid * PSTR;
    int k = 0;
#pragma unroll
    for (int i = 0; i < 16; i++)
#pragma unroll
      for (int j = i + 1; j < 16; j++)
        pr[k++] = (u[i] * v[j] - u[j] * v[i]) * inv;
  }
  __syncthreads();

  // phase 3: GEMM1 (K=120) + gelu + causal-masked average -> q
  float* qb = ZQ;
#pragma unroll 1
  for (int nt = wid * 4; nt < wid * 4 + 4; ++nt) {
    const int col = nt * 16 + ln;
    const float b1 = g1b[col];
    v8f qacc = {0.f, 0.f, 0.f, 0.f, 0.f, 0.f, 0.f, 0.f};
#pragma unroll 1
    for (int d = 0; d < 6; ++d) {
      v8f acc = {0.f, 0.f, 0.f, 0.f, 0.f, 0.f, 0.f, 0.f};
      const float* pa = P + (d * 16 + ln) * PSTR + koff;
#pragma unroll 2
      for (int kk = 0; kk < 30; ++kk) {
        int k0 = kk * 4;
        v2f a; a.x = pa[k0]; a.y = pa[k0 + 1];
        v2f b;
        b.x = g1w[(k0 + koff) * 256 + col];
        b.y = g1w[(k0 + koff + 1) * 256 + col];
        acc = WMMA_F32(a, b, acc);
      }
      int delta = 1 << d;
#pragma unroll
      for (int i = 0; i < 8; i++) {
        int m = i + 8 * hi;
        float msk = ((l0 + m) >= delta) ? 1.0f : 0.0f;
        qacc[i] += msk * gelu_exact(acc[i] + b1);
      }
    }
#pragma unroll
    for (int i = 0; i < 8; i++) {
      int m = i + 8 * hi;
      int l = l0 + m;
      float cnt = (float)((l >= 1) + (l >= 2) + (l >= 4) + (l >= 8) +
                          (l >= 16) + (l >= 32));
      qb[m * QSTR + col] = qacc[i] / fmaxf(cnt, 1.0f);
    }
  }
  __syncthreads();

  // phase 4: GEMM2 (16x256)@(256x1024) -> g
#pragma unroll 1
  for (int j = 0; j < 16; ++j) {
    const int col = (wid * 16 + j) * 16 + ln;
    v8f acc = {0.f, 0.f, 0.f, 0.f, 0.f, 0.f, 0.f, 0.f};
    const float* qa = qb + ln * QSTR + koff;
#pragma unroll 2
    for (int kk = 0; kk < 64; ++kk) {
      int k0 = kk * 4;
      v2f a; a.x = qa[k0]; a.y = qa[k0 + 1];
      v2f b;
      b.x = g2w[(k0 + koff) * 1024 + col];
      b.y = g2w[(k0 + koff + 1) * 1024 + col];
      acc = WMMA_F32(a, b, acc);
    }
    const float b2 = g2b[col];
#pragma unroll
    for (int i = 0; i < 8; i++) {
      int m = i + 8 * hi;
      float bs = ((l0 + m) >= 1) ? 1.0f : 0.0f;  // count/max(count,1)
      g[(size_t)(t0 + m) * 1024 + col] = acc[i] + bs * b2;
    }
  }
}

// ---------------------------------------------------------------------------
// Kernel 3: gate GEMM (M=16384, N=1024, K=2048) + sigmoid blend.
// Block = 64 tokens x 64 cols, 4 waves; each wave: 4 M-tiles x 1 N-tile.
// A tile (64 x 64 f32, h rows then g rows) DMA'd into LDS by the TDM,
// double-buffered, overlapped with WMMA compute.
// ---------------------------------------------------------------------------
#define ASTR 68  // 64 data DWORDs + 4 pad DWORDs (written by TDM padding)

__global__ __launch_bounds__(128) void gatek(const float* __restrict__ h,
                                             const float* __restrict__ g,
                                             const float* __restrict__ gw,
                                             const float* __restrict__ gb,
                                             float* __restrict__ out) {
  __shared__ float As[2 * 64 * ASTR];  // 34816 B double buffer
  const int tid  = threadIdx.x;
  const int lane = tid & 31;
  const int wid  = tid >> 5;
  const int ln   = lane & 15;
  const int hi   = lane >> 4;
  const int koff = hi * 2;
  const int t0  = blockIdx.x * 64;
  const int col = blockIdx.y * 64 + wid * 16 + ln;

  v8f acc[4];
#pragma unroll
  for (int mt = 0; mt < 4; mt++)
    acc[mt] = (v8f){0.f, 0.f, 0.f, 0.f, 0.f, 0.f, 0.f, 0.f};

  if (wid == 0)
    tdm_load_64x64(h + (size_t)t0 * 1024, (unsigned)(uintptr_t)&As[0]);

  // K = 2048: chunks 0..15 read h rows, 16..31 read g rows; gate_w row = c*64+k
#pragma unroll 1
  for (int c = 0; c < 32; ++c) {
    if (wid == 0) __builtin_amdgcn_s_wait_tensorcnt(0);
    __syncthreads();
    if (wid == 0 && c + 1 < 32) {
      int cn = c + 1;
      const float* base = (cn < 16) ? h : g;
      const float* gsrc = base + (size_t)t0 * 1024 + (size_t)(cn & 15) * 64;
      tdm_load_64x64(gsrc, (unsigned)(uintptr_t)&As[(cn & 1) * 64 * ASTR]);
    }
    const float* Ac = &As[(c & 1) * 64 * ASTR];
#pragma unroll 2
    for (int kk = 0; kk < 16; ++kk) {
      int krow = c * 64 + kk * 4 + koff;
      v2f b;
      b.x = gw[(size_t)krow * 1024 + col];
      b.y = gw[(size_t)(krow + 1) * 1024 + col];
      const float* arow = Ac + kk * 4 + koff;
#pragma unroll
      for (int mt = 0; mt < 4; ++mt) {
        const float* ap = arow + (mt * 16 + ln) * ASTR;
        v2f a; a.x = ap[0]; a.y = ap[1];
        acc[mt] = WMMA_F32(a, b, acc[mt]);
      }
    }
    __syncthreads();
  }

  const float bias = gb[col];
#pragma unroll
  for (int mt = 0; mt < 4; ++mt)
#pragma unroll
    for (int i = 0; i < 8; i++) {
      int row = t0 + mt * 16 + i + 8 * hi;
      float x = acc[mt][i] + bias;
      float alpha = 1.0f / (1.0f + expf(-x));
      size_t idx = (size_t)row * 1024 + col;
      float hv = h[idx], gv = g[idx];
      out[idx] = alpha * hv + (1.0f - alpha) * gv;
    }
}

// ---------------------------------------------------------------------------
extern "C" void kernel_launch(void* const* d_in, const int* in_sizes, int n_in,
                              void* d_out, int out_size, void* d_ws, size_t ws_size,
                              hipStream_t stream) {
  const float* h   = (const float*)d_in[0];
  const float* rw  = (const float*)d_in[1];
  const float* rb  = (const float*)d_in[2];
  const float* g1w = (const float*)d_in[3];
  const float* g1b = (const float*)d_in[4];
  const float* g2w = (const float*)d_in[5];
  const float* g2b = (const float*)d_in[6];
  const float* gw  = (const float*)d_in[7];
  const float* gbv = (const float*)d_in[8];
  float* out = (float*)d_out;

  const int NTOK = 4 * 4096;         // B*L
  float* z = (float*)d_ws;           // NTOK*16 floats (1 MB)
  float* g = z + (size_t)NTOK * 16;  // NTOK*1024 floats (64 MB)

  zk<<<NTOK / 8, 256, 0, stream>>>(h, rw, rb, z, NTOK);
  geomk<<<NTOK / 16, 128, 0, stream>>>(z, g1w, g1b, g2w, g2b, g);
  gatek<<<dim3(NTOK / 64, 16), 128, 0, stream>>>(h, g, gw, gbv, out);
}